// RelPosSelfAttention_5360119185814
// MI455X (gfx1250) — compile-verified
//
#include <hip/hip_runtime.h>
#include <hip/hip_bf16.h>
#include <math.h>

// ---------------------------------------------------------------------------
// RelPos self-attention for MI455X (gfx1250, wave32).
// fp32 WMMA (V_WMMA_F32_16X16X4_F32) for all GEMM-shaped math (matches fp32
// reference). Weight panels staged into LDS via the Tensor Data Mover
// (tensor_load_to_lds) with TDM padding to avoid LDS bank conflicts.
// Whole working set except attn/bias (134MB each) is L2-resident (192MB).
// ---------------------------------------------------------------------------

typedef __attribute__((ext_vector_type(2))) float v2f;
typedef __attribute__((ext_vector_type(8))) float v8f;
typedef __attribute__((ext_vector_type(4))) int   v4i;
typedef __attribute__((ext_vector_type(8))) int   v8i;

#define Bsz   4
#define Ntok  1024
#define Dmod  512
#define Hn    8
#define HD    64
#define NS    1

// D = A*B + C, fp32 WMMA 16x16x4 (8 args: neg_a,A,neg_b,B,c_mod,C,reuse_a,reuse_b)
__device__ __forceinline__ v8f wmma_f32(v2f a, v2f b, v8f c) {
  return __builtin_amdgcn_wmma_f32_16x16x4_f32(false, a, false, b, (short)0, c, false, false);
}

// ---------------------------------------------------------------------------
// TDM staging: copy rows [row0, row0+64) x [0,512) of W (row-major, fp32) into
// LDS with 2-DWORD padding after every 256 DWORDs (pad_interval=7, pad_amount=1)
// -> effective row stride 516 floats; element (c,k) at c*516 + k + 2*(k>=256).
// Issued by wave 0 only; completion via TENSORcnt.
// ---------------------------------------------------------------------------
__device__ __forceinline__ void tdm_stage_panel(const float* gsrc, unsigned lds_byte_addr) {
  unsigned long long ga = (unsigned long long)(uintptr_t)gsrc;
  v4i g0;
  g0[0] = 1;                                   // count=1 valid descriptor
  g0[1] = (int)lds_byte_addr;                  // lds_addr
  g0[2] = (int)(ga & 0xFFFFFFFFull);           // global_addr[31:0]
  g0[3] = (int)((ga >> 32) & 0x01FFFFFFull) | (2 << 30);  // addr[56:32] | type=2
  v8i g1;
  g1[0] = (2 << 16)        // data_size = 4 bytes
        | (1 << 20)        // pad_enable
        | (7 << 22)        // pad_interval: every 256 DWORDs
        | (1 << 25);       // pad_amount: 2 DWORDs
  g1[1] = (512 << 16);     // tensor_dim0[15:0] = 512 (elements per row)
  g1[2] = (64 << 16);      // tensor_dim0[31:16]=0 | tensor_dim1[15:0] = 64 rows
  g1[3] = (512 << 16);     // tensor_dim1[31:16]=0 | tile_dim0 = 512
  g1[4] = 64;              // tile_dim1 = 64 rows (tile_dim2 = 0)
  g1[5] = 512;             // tensor_dim0_stride[31:0] = 512 elements
  g1[6] = 0;               // stride hi / dim1_stride lo
  g1[7] = 0;
  asm volatile("tensor_load_to_lds %0, %1" :: "s"(g0), "s"(g1) : "memory");
  __builtin_amdgcn_s_wait_tensorcnt(0);
}

// ---------------------------------------------------------------------------
// GEMM: C[M,Nout] = A[M,512] * W[Nout,512]^T (+ biasv). K fixed at 512.
// Block = 256 threads = 8 waves arranged 4(M) x 2(N); block tile 128x64.
// Each wave register-blocks a 32x32 output (4 fp32-WMMA accumulators), so A
// and B fragments are each reused twice per K-step (4 WMMA per 4 frag loads).
// Weight panel (64 rows x 512) staged once via TDM (132KB LDS, padded).
// ---------------------------------------------------------------------------
__global__ void __launch_bounds__(256)
gemm_wmma_f32_k512(const float* __restrict__ A,
                   const float* __restrict__ W,
                   float* __restrict__ C,
                   const float* __restrict__ biasv,
                   int M, int Nout)
{
  extern __shared__ float lds[];          // 64 * 516 floats (TDM-padded panel)
  const int LDB = 516;
  const int n0 = blockIdx.x * 64;
  const int m0 = blockIdx.y * 128;

  const int lane = threadIdx.x & 31;
  const int wave = threadIdx.x >> 5;

  if (wave == 0) {
    // dynamic LDS begins at groupstaticsize offset
    tdm_stage_panel(W + (size_t)n0 * 512, __builtin_amdgcn_groupstaticsize());
  }
  __syncthreads();

  const int mq = wave >> 1;               // 0..3 : which 32-row strip
  const int nq = wave & 1;                // 0..1 : which 32-col strip
  const int r  = lane & 15;               // tile row / col index
  const int kh = (lane >> 4) * 2;         // K sub-pair this half-wave owns
  const int mrow = m0 + mq * 32;
  const int c0 = nq * 32 + r;             // panel columns for B fragments
  const int c1 = c0 + 16;

  const float* arow0 = A + (size_t)(mrow + r) * 512;
  const float* arow1 = arow0 + (size_t)16 * 512;
  __builtin_prefetch(arow0, 0, 0);        // global_prefetch_b8
  __builtin_prefetch(arow1, 0, 0);

  v8f acc00 = {}, acc01 = {}, acc10 = {}, acc11 = {};
#pragma unroll 4
  for (int k0 = 0; k0 < 512; k0 += 4) {
    const int kk  = k0 + kh;
    const int off = kk + ((kk >> 8) << 1);   // TDM pad: +2 DWORDs past k=255
    v2f a0, a1, b0, b1;
    a0.x = arow0[kk];          a0.y = arow0[kk + 1];
    a1.x = arow1[kk];          a1.y = arow1[kk + 1];
    b0.x = lds[c0 * LDB + off]; b0.y = lds[c0 * LDB + off + 1];
    b1.x = lds[c1 * LDB + off]; b1.y = lds[c1 * LDB + off + 1];
    acc00 = wmma_f32(a0, b0, acc00);
    acc01 = wmma_f32(a0, b1, acc01);
    acc10 = wmma_f32(a1, b0, acc10);
    acc11 = wmma_f32(a1, b1, acc11);
  }

  // C layout per tile: lanes 0-15 -> M=vg, lanes 16-31 -> M=vg+8; col = lane&15
  const int mb = (lane >> 4) * 8;
  const float bv0 = biasv ? biasv[n0 + nq * 32 + r]      : 0.f;
  const float bv1 = biasv ? biasv[n0 + nq * 32 + 16 + r] : 0.f;
#pragma unroll
  for (int vg = 0; vg < 8; ++vg) {
    const size_t r0 = (size_t)(mrow + mb + vg) * Nout;
    const size_t r1 = (size_t)(mrow + 16 + mb + vg) * Nout;
    C[r0 + n0 + nq * 32 + r]      = acc00[vg] + bv0;
    C[r0 + n0 + nq * 32 + 16 + r] = acc01[vg] + bv1;
    C[r1 + n0 + nq * 32 + r]      = acc10[vg] + bv0;
    C[r1 + n0 + nq * 32 + 16 + r] = acc11[vg] + bv1;
  }
}

// ---------------------------------------------------------------------------
// Bias MLP: one thread per (b,i,j). Layer-1 (4->32, exact gelu) computed once
// and shared across all 8 heads; stores alpha*bias*valid, layout [B,N,N,H].
// ---------------------------------------------------------------------------
__global__ void bias_mlp(const float* __restrict__ coords,
                         const float* __restrict__ W1, const float* __restrict__ b1,
                         const float* __restrict__ W2, const float* __restrict__ b2,
                         const float* __restrict__ alpha_p,
                         float* __restrict__ biasbuf)
{
  const size_t idx = (size_t)blockIdx.x * 256 + threadIdx.x;   // B*N*N threads
  const int b   = (int)(idx >> 20);
  const int rem = (int)(idx & ((1u << 20) - 1));
  const int i   = rem >> 10;
  const int j   = rem & 1023;

  const float* ci = coords + ((size_t)b * Ntok + i) * 3;
  const float* cj = coords + ((size_t)b * Ntok + j) * 3;
  const float d0 = ci[0] - cj[0];
  const float d1 = ci[1] - cj[1];
  const float d2 = ci[2] - cj[2];
  const float dist = sqrtf(d0 * d0 + d1 * d1 + d2 * d2);

  const float alpha = alpha_p[0];
  const bool valid  = (i >= NS) && (j >= NS);

  float out[Hn];
#pragma unroll
  for (int h = 0; h < Hn; ++h) out[h] = b2[h];

#pragma unroll 4
  for (int u = 0; u < 32; ++u) {
    float z = W1[u * 4 + 0] * d0 + W1[u * 4 + 1] * d1 +
              W1[u * 4 + 2] * d2 + W1[u * 4 + 3] * dist + b1[u];
    float g = 0.5f * z * (1.f + erff(z * 0.70710678118654752f));  // exact gelu
#pragma unroll
    for (int h = 0; h < Hn; ++h) out[h] += W2[h * 32 + u] * g;
  }

  float* dst = biasbuf + idx * Hn;
#pragma unroll
  for (int h = 0; h < Hn; ++h) dst[h] = valid ? alpha * out[h] : 0.f;
}

// ---------------------------------------------------------------------------
// Fused attention per (b, h, 16-row i-tile):
//   phase 1: logits = (q*scale)·k^T (WMMA) + bias, key-padding mask -> LDS
//   phase 2: softmax over 1024-wide rows (wave32 shfl reductions), write attn
//   phase 3: O = attn·V via WMMA, K split 2-ways across waves, LDS reduce
// ---------------------------------------------------------------------------
__global__ void attn_fused(const float* __restrict__ qkv,       // [B*N,1536]
                           const float* __restrict__ biasbuf,   // [B,N,N,H]
                           const unsigned char* __restrict__ kpm, // [B,N] bool
                           float* __restrict__ attn_out,        // [B,H,N,N]
                           float* __restrict__ obuf)            // [B*N,512]
{
  extern __shared__ float lds[];
  const int LQ = 68;     // q stride (64 + 4 pad)
  const int LL = 1032;   // logits stride (1024 + 8 pad)
  float* q_s  = lds;                         // 16*68   = 1088
  float* lg   = lds + 16 * LQ;               // 16*1032 = 16512
  float* part = lg + 16 * LL;                // 4*16*16 = 1024

  const int i0 = blockIdx.x * 16;
  const int h  = blockIdx.y;
  const int b  = blockIdx.z;
  const int lane = threadIdx.x & 31;
  const int wave = threadIdx.x >> 5;
  const float scale = 0.125f;                // HD^-0.5 = 1/8

  // stage q tile, pre-scaled
  for (int idx = threadIdx.x; idx < 16 * HD / 4; idx += 256) {
    int r  = idx >> 4;
    int d4 = (idx & 15) << 2;
    const float4 v = *(const float4*)&qkv[((size_t)(b * Ntok + i0 + r)) * 1536 + h * HD + d4];
    q_s[r * LQ + d4 + 0] = v.x * scale;
    q_s[r * LQ + d4 + 1] = v.y * scale;
    q_s[r * LQ + d4 + 2] = v.z * scale;
    q_s[r * LQ + d4 + 3] = v.w * scale;
  }
  __syncthreads();

  const int r  = lane & 15;
  const int kh = (lane >> 4) * 2;
  const int mb = (lane >> 4) * 8;
  const float NEG_INF = -__builtin_inff();

  // ---- phase 1: logits ----
  for (int t = 0; t < 8; ++t) {
    const int j0 = (wave + 8 * t) * 16;
    v8f acc = {};
    const float* kbase = qkv + (size_t)(b * Ntok + j0 + r) * 1536 + Dmod + h * HD;
    __builtin_prefetch(kbase, 0, 0);
#pragma unroll 4
    for (int k0 = 0; k0 < HD; k0 += 4) {
      v2f a, bb;
      a.x  = q_s[r * LQ + k0 + kh];
      a.y  = q_s[r * LQ + k0 + kh + 1];
      bb.x = kbase[k0 + kh];
      bb.y = kbase[k0 + kh + 1];
      acc = wmma_f32(a, bb, acc);
    }
    const int j = j0 + r;
    const bool pad = kpm[b * Ntok + j] != 0;
#pragma unroll
    for (int vg = 0; vg < 8; ++vg) {
      const int i = i0 + mb + vg;
      float v = acc[vg] + biasbuf[(((size_t)b * Ntok + i) * Ntok + j) * Hn + h];
      lg[(mb + vg) * LL + j] = pad ? NEG_INF : v;
    }
  }
  __syncthreads();

  // ---- phase 2: softmax (wave w handles rows 2w, 2w+1) ----
  for (int rr = wave * 2; rr < wave * 2 + 2; ++rr) {
    float* row = lg + rr * LL;
    float m = NEG_INF;
    for (int j = lane; j < Ntok; j += 32) m = fmaxf(m, row[j]);
#pragma unroll
    for (int off = 16; off; off >>= 1) m = fmaxf(m, __shfl_xor(m, off, 32));
    float s = 0.f;
    for (int j = lane; j < Ntok; j += 32) { float e = __expf(row[j] - m); row[j] = e; s += e; }
#pragma unroll
    for (int off = 16; off; off >>= 1) s += __shfl_xor(s, off, 32);
    const float inv = 1.f / s;
    for (int j = lane; j < Ntok; j += 32) row[j] *= inv;
  }
  __syncthreads();

  // write attn tile to global (coalesced float4)
  {
    float* dst = attn_out + (((size_t)(b * Hn + h) * Ntok) + i0) * Ntok;
    for (int idx = threadIdx.x; idx < 16 * Ntok / 4; idx += 256) {
      int r2 = idx >> 8;
      int c4 = (idx & 255) << 2;
      float4 v;
      v.x = lg[r2 * LL + c4 + 0];
      v.y = lg[r2 * LL + c4 + 1];
      v.z = lg[r2 * LL + c4 + 2];
      v.w = lg[r2 * LL + c4 + 3];
      *(float4*)&dst[r2 * Ntok + c4] = v;
    }
  }

  // ---- phase 3: O = attn * V; wave = (dtile in 0..3, K half in 0..1) ----
  const int dt   = wave & 3;
  const int ksec = wave >> 2;
  {
    v8f acc = {};
    const int kstart = ksec * 512;
#pragma unroll 4
    for (int k0 = kstart; k0 < kstart + 512; k0 += 4) {
      v2f a, bb;
      a.x = lg[r * LL + k0 + kh];
      a.y = lg[r * LL + k0 + kh + 1];
      const float* vb = qkv + (size_t)(b * Ntok + k0 + kh) * 1536 + 2 * Dmod + h * HD + dt * 16 + r;
      bb.x = vb[0];
      bb.y = vb[1536];
      acc = wmma_f32(a, bb, acc);
    }
    if (ksec == 1) {
#pragma unroll
      for (int vg = 0; vg < 8; ++vg) part[(dt * 16 + mb + vg) * 16 + r] = acc[vg];
    }
    __syncthreads();
    if (ksec == 0) {
#pragma unroll
      for (int vg = 0; vg < 8; ++vg) {
        float sum = acc[vg] + part[(dt * 16 + mb + vg) * 16 + r];
        obuf[(size_t)(b * Ntok + i0 + mb + vg) * Dmod + h * HD + dt * 16 + r] = sum;
      }
    }
  }
}

// ---------------------------------------------------------------------------
extern "C" void kernel_launch(void* const* d_in, const int* in_sizes, int n_in,
                              void* d_out, int out_size, void* d_ws, size_t ws_size,
                              hipStream_t stream) {
  const float* x      = (const float*)d_in[0];
  const float* coords = (const float*)d_in[1];
  const unsigned char* kpm = (const unsigned char*)d_in[2]; // bool mask
  const float* alpha  = (const float*)d_in[3];
  const float* W1     = (const float*)d_in[4];
  const float* b1     = (const float*)d_in[5];
  const float* W2     = (const float*)d_in[6];
  const float* b2     = (const float*)d_in[7];
  const float* Wqkv   = (const float*)d_in[8];
  const float* Wout   = (const float*)d_in[9];
  const float* bout   = (const float*)d_in[10];

  float* out  = (float*)d_out;                                 // [B,N,D]
  float* attn = out + (size_t)Bsz * Ntok * Dmod;               // [B,H,N,N]

  char* ws = (char*)d_ws;
  float* qkv     = (float*)ws;                                 // 4096*1536*4 = 25.2 MB
  float* obuf    = (float*)(ws + (size_t)25165824);            // 4096*512*4  =  8.4 MB
  float* biasbuf = (float*)(ws + (size_t)25165824 + 8388608);  // 4M*8*4      = 134  MB

  const size_t gemm_lds = (size_t)64 * 516 * sizeof(float);    // 132 KB (TDM-padded)
  const size_t attn_lds = (size_t)(16 * 68 + 16 * 1032 + 1024) * sizeof(float); // ~74.5 KB

  // 1) qkv = x @ Wqkv^T
  gemm_wmma_f32_k512<<<dim3(1536 / 64, 4096 / 128), 256, gemm_lds, stream>>>(
      x, Wqkv, qkv, nullptr, 4096, 1536);

  // 2) bias (all heads at once), already *alpha and valid-masked
  bias_mlp<<<(Bsz * Ntok * Ntok) / 256, 256, 0, stream>>>(
      coords, W1, b1, W2, b2, alpha, biasbuf);

  // 3) fused attention (writes attn to d_out, O to obuf)
  attn_fused<<<dim3(Ntok / 16, Hn, Bsz), 256, attn_lds, stream>>>(
      qkv, biasbuf, kpm, attn, obuf);

  // 4) out = O @ Wout^T + bout
  gemm_wmma_f32_k512<<<dim3(Dmod / 64, 4096 / 128), 256, gemm_lds, stream>>>(
      obuf, Wout, out, bout, 4096, Dmod);
}